// DynamicMaskHead_32521492365761
// MI455X (gfx1250) — compile-verified
//
#include <hip/hip_runtime.h>

typedef __attribute__((ext_vector_type(2))) float v2f;
typedef __attribute__((ext_vector_type(4))) float v4f;
typedef __attribute__((ext_vector_type(8))) float v8f;

#define C_IN      8
#define CH        8
#define H_IMG     100
#define W_IMG     136
#define NPIX      (H_IMG * W_IMG)      // 13600
#define NPARAM    232                  // 80 + 64 + 64 + 8 + 8 + 8
#define NPAR_PAD  240                  // + 8-float zero pad for hi-lane bias reads
#define OFF_W1    0
#define OFF_W2    80
#define OFF_W3    144
#define OFF_B1    208
#define OFF_B2    216
#define OFF_B3    224
#define OFF_ZPAD  232

#define TILE_P    512                  // pixels per block chunk
#define LDS_ROW   528                  // row stride: %64 == 16 bank shift, %4 == 0
#define NCHUNK    ((NPIX + TILE_P - 1) / TILE_P)   // 27

// D = A(16x4) * B(4x16) + C, fp32 WMMA, wave32.
static __device__ __forceinline__ v8f wmma4(v2f a, v2f b, v8f c) {
    return __builtin_amdgcn_wmma_f32_16x16x4_f32(
        false, a, false, b, (short)0, c, false, false);
}

// single-instruction ReLU: v_med3_f32(x, 0, +MAX)
static __device__ __forceinline__ float relu1(float x) {
    return __builtin_amdgcn_fmed3f(x, 0.0f, 3.402823466e38f);
}

// opacity barrier: pins a value in a VGPR so it can't be rematerialized/reloaded
static __device__ __forceinline__ void pin(float& x) {
    asm volatile("" : "+v"(x));
}

__global__ __launch_bounds__(256)
void dyn_mask_head_wmma(const float* __restrict__ mask_feats,
                        const float* __restrict__ params,
                        const float* __restrict__ locs,
                        const int*   __restrict__ im_inds,
                        const int*   __restrict__ fpn_levels,
                        float*       __restrict__ out) {
    __shared__ float s_feats[C_IN * LDS_ROW];
    __shared__ float s_out[CH * LDS_ROW];
    __shared__ float s_par[NPAR_PAD];

    const int  inst = blockIdx.y;
    const int  p0   = blockIdx.x * TILE_P;
    const int  tid  = threadIdx.x;
    const int  lane = tid & 31;
    const int  wave = tid >> 5;          // 8 waves; wave == channel in the copy passes
    const int  mrow = lane & 15;
    const bool hi   = lane >= 16;        // A/B: K=2,3 half; C/D: M=8..15 half

    // ---- stage per-instance params (+ zero pad) ----
    if (tid < NPAR_PAD)
        s_par[tid] = (tid < NPARAM) ? params[(size_t)inst * NPARAM + tid] : 0.0f;

    // per-instance scalars (block-uniform)
    const float locx = locs[2 * inst + 0];
    const float locy = locs[2 * inst + 1];
    const int   lvl  = fpn_levels[inst];
    // SOI = 64 * 2^lvl (power of two) -> exact reciprocal via exponent bits
    union { unsigned u; float f; } rcp; rcp.u = (unsigned)(121 - lvl) << 23;
    const float inv_soi = rcp.f;                           // 2^-(6+lvl)
    const float* fbase  = mask_feats + (size_t)im_inds[inst] * (C_IN * NPIX);

    // ---- stage feature tile: wave w copies channel w, 4x 128-px b128 blocks ----
    {
        const float* src = fbase + (size_t)wave * NPIX;
        float*       dst = &s_feats[wave * LDS_ROW];
#pragma unroll
        for (int b = 0; b < 4; ++b) {
            const int o4 = b * 128 + lane * 4;
            const int gp = p0 + o4;
            v4f v = {0.f, 0.f, 0.f, 0.f};
            if (gp + 3 < NPIX) {
                v = *(const v4f*)(src + gp);
            } else {
#pragma unroll
                for (int k = 0; k < 4; ++k)
                    if (gp + k < NPIX) v[k] = src[gp + k];
            }
            *(v4f*)(dst + o4) = v;
        }
    }
    __syncthreads();

    // ---- wave-invariant A matrices (A layout: lane = m | (k>=2)<<4, vgpr = k&1) ----
    const int kb = hi ? 2 : 0;
    auto wA = [&](int off, int ncol, int i) -> float {
        return (mrow < CH && i < ncol) ? s_par[off + mrow * ncol + i] : 0.0f;
    };
    v2f a10, a11, a12, a20, a21, a30, a31;
    a10[0] = wA(OFF_W1, 10, 0 + kb); a10[1] = wA(OFF_W1, 10, 1 + kb);
    a11[0] = wA(OFF_W1, 10, 4 + kb); a11[1] = wA(OFF_W1, 10, 5 + kb);
    a12[0] = wA(OFF_W1, 10, 8 + kb); a12[1] = wA(OFF_W1, 10, 9 + kb);  // cols 10,11 == 0
    a20[0] = wA(OFF_W2,  8, 0 + kb); a20[1] = wA(OFF_W2,  8, 1 + kb);
    a21[0] = wA(OFF_W2,  8, 4 + kb); a21[1] = wA(OFF_W2,  8, 5 + kb);
    a30[0] = wA(OFF_W3,  8, 0 + kb); a30[1] = wA(OFF_W3,  8, 1 + kb);
    a31[0] = wA(OFF_W3,  8, 4 + kb); a31[1] = wA(OFF_W3,  8, 5 + kb);

    // biases through the initial C operand: C[m][n] = b[m] for m<8, 0 for m>=8.
    // hi lanes read the zero pad; values pinned in VGPRs (no per-tile reload/remat).
    v8f c1, c2, c3;
    {
        const float* pb1 = &s_par[hi ? OFF_ZPAD : OFF_B1];
        const float* pb2 = &s_par[hi ? OFF_ZPAD : OFF_B2];
        const float* pb3 = &s_par[hi ? OFF_ZPAD : OFF_B3];
        const v4f b1l = *(const v4f*)pb1,       b1h = *(const v4f*)(pb1 + 4);
        const v4f b2l = *(const v4f*)pb2,       b2h = *(const v4f*)(pb2 + 4);
        const v4f b3l = *(const v4f*)pb3,       b3h = *(const v4f*)(pb3 + 4);
#pragma unroll
        for (int r = 0; r < 4; ++r) {
            c1[r] = b1l[r]; c1[r + 4] = b1h[r];
            c2[r] = b2l[r]; c2[r + 4] = b2h[r];
            c3[r] = b3l[r]; c3[r + 4] = b3h[r];
        }
#pragma unroll
        for (int r = 0; r < 8; ++r) {
            float t;
            t = c1[r]; pin(t); c1[r] = t;
            t = c2[r]; pin(t); c2[r] = t;
            t = c3[r]; pin(t); c3[r] = t;
        }
    }

    // ---- per-lane base pointers for the B-operand loads (loop-invariant) ----
    // L0/L1: hi -> ch0/ch1 (chunk0 K=2,3); lo -> ch6/ch7 (chunk2 K=0,1)
    // L2/L3: hi -> ch4/ch5 (chunk1 K=2,3); lo -> ch2/ch3 (chunk1 K=0,1)
    const int colbase = wave * 16 + mrow;
    const float* fp0 = &s_feats[(hi ? 0 : 6) * LDS_ROW + colbase];
    const float* fp1 = &s_feats[(hi ? 1 : 7) * LDS_ROW + colbase];
    const float* fp2 = &s_feats[(hi ? 4 : 2) * LDS_ROW + colbase];
    const float* fp3 = &s_feats[(hi ? 5 : 3) * LDS_ROW + colbase];
    float*       sop = &s_out[colbase];

    // incremental pixel coordinates for this lane's column (float counters, exact)
    const int gp0 = p0 + colbase;
    int       py0 = gp0 / W_IMG;
    int       px  = gp0 - py0 * W_IMG;
    float     fpx = (float)px;
    float     fpy = (float)py0;
    const float cx = (locx - 4.0f) * inv_soi;    // rx = cx + s8*px
    const float cy = (locy - 4.0f) * inv_soi;
    const float s8 = -8.0f * inv_soi;

    // ---- 4 fully-unrolled 16-pixel tiles per wave (EXEC all-ones at every WMMA) ----
#pragma unroll
    for (int u = 0; u < 4; ++u) {
        const float L0 = fp0[u * 128];
        const float L1 = fp1[u * 128];
        const float L2 = fp2[u * 128];
        const float L3 = fp3[u * 128];

        const float rx = fmaf(s8, fpx, cx);
        const float ry = fmaf(s8, fpy, cy);

        v2f B0, B1, B2;
        B0[0] = hi ? L0 : rx;   B0[1] = hi ? L1 : ry;
        B1[0] = L2;             B1[1] = L3;
        B2[0] = L0;             B2[1] = L1;      // hi half hits zero A columns (10,11)

        // layer 1: K = {rx,ry,f0,f1 | f2..f5 | f6,f7,-,-}
        v8f x = wmma4(a10, B0, c1);
        x = wmma4(a11, B1, x);
        x = wmma4(a12, B2, x);

        // D -> B layout conversion, ReLU applied once per surviving value
        v2f B20, B21;
        {
            const float t0 = __shfl(x[2], mrow, 32);
            const float t1 = __shfl(x[3], mrow, 32);
            const float t2 = __shfl(x[6], mrow, 32);
            const float t3 = __shfl(x[7], mrow, 32);
            B20[0] = relu1(hi ? t0 : x[0]);  B20[1] = relu1(hi ? t1 : x[1]);
            B21[0] = relu1(hi ? t2 : x[4]);  B21[1] = relu1(hi ? t3 : x[5]);
        }

        v8f y = wmma4(a20, B20, c2);
        y = wmma4(a21, B21, y);

        v2f B30, B31;
        {
            const float t0 = __shfl(y[2], mrow, 32);
            const float t1 = __shfl(y[3], mrow, 32);
            const float t2 = __shfl(y[6], mrow, 32);
            const float t3 = __shfl(y[7], mrow, 32);
            B30[0] = relu1(hi ? t0 : y[0]);  B30[1] = relu1(hi ? t1 : y[1]);
            B31[0] = relu1(hi ? t2 : y[4]);  B31[1] = relu1(hi ? t3 : y[5]);
        }

        v8f z = wmma4(a30, B30, c3);
        z = wmma4(a31, B31, z);

        // park results: row m is in vgpr m on lanes 0-15 (final ReLU on the 8 stored values)
        if (!hi) {
#pragma unroll
            for (int r = 0; r < 8; ++r) sop[r * LDS_ROW + u * 128] = relu1(z[r]);
        }

        // advance this lane's column by 128 pixels
        px += 128;
        const bool w = px >= W_IMG;
        px  = w ? px - W_IMG : px;
        fpx = w ? fpx - 8.0f : fpx + 128.0f;     // (128 - 136) vs +128
        fpy = w ? fpy + 1.0f : fpy;
    }
    __syncthreads();

    // ---- coalesced NT store pass: wave w stores channel w ----
    {
        float*       dst = out + (size_t)inst * CH * NPIX + (size_t)wave * NPIX;
        const float* src = &s_out[wave * LDS_ROW];
#pragma unroll
        for (int b = 0; b < 4; ++b) {
            const int o4 = b * 128 + lane * 4;
            const int gp = p0 + o4;
            if (gp + 3 < NPIX) {
                __builtin_nontemporal_store(*(const v4f*)(src + o4), (v4f*)(dst + gp));
            } else {
#pragma unroll
                for (int k = 0; k < 4; ++k)
                    if (gp + k < NPIX)
                        __builtin_nontemporal_store(src[o4 + k], dst + gp + k);
            }
        }
    }
}

extern "C" void kernel_launch(void* const* d_in, const int* in_sizes, int n_in,
                              void* d_out, int out_size, void* d_ws, size_t ws_size,
                              hipStream_t stream) {
    const float* mask_feats = (const float*)d_in[0];
    const float* params     = (const float*)d_in[1];
    const float* locs       = (const float*)d_in[2];
    const int*   im_inds    = (const int*)d_in[3];
    const int*   fpn_levels = (const int*)d_in[4];
    float*       out        = (float*)d_out;

    const int n_inst = in_sizes[1] / NPARAM;   // 400
    if (n_inst <= 0) return;

    dim3 grid(NCHUNK, n_inst);
    dim3 block(256);
    dyn_mask_head_wmma<<<grid, block, 0, stream>>>(mask_feats, params, locs,
                                                   im_inds, fpn_levels, out);
}